// VectorQuantizer_2027224564516
// MI455X (gfx1250) — compile-verified
//
#include <hip/hip_runtime.h>
#include <hip/hip_bf16.h>

// -------- CDNA5 (gfx1250) VQ-VAE vector quantizer --------
// dims (fixed by reference):
//   latents  [B=64, D=64, H=32, W=32]  fp32, NCHW
//   embedding[K=1024, D=64]            fp32
//   out = quant NCHW (4194304 floats) ++ vq_loss (1 float)
//
// Distance GEMM done with V_WMMA_F32_16X16X4_F32 (exact fp32 -> argmin parity).

typedef __attribute__((ext_vector_type(2))) float v2f;
typedef __attribute__((ext_vector_type(8))) float v8f;

#define NB   64
#define DD   64
#define HW   1024          // H*W
#define KK   1024
#define NN   65536         // NB*HW
#define NELEM 4194304      // NN*DD
#define LDS_STRIDE 66      // padded row stride (floats): conflict-free, 8B-aligned pairs

// ---------------- kernel 0: ||e_k||^2 ----------------
__global__ void __launch_bounds__(256) vq_enorm_kernel(const float* __restrict__ E,
                                                       float* __restrict__ enorm) {
  int k = blockIdx.x * 256 + threadIdx.x;     // grid = 4 blocks -> k in [0,1024)
  const float4* row = (const float4*)(E + (size_t)k * DD);
  float s = 0.f;
#pragma unroll
  for (int t = 0; t < DD / 4; ++t) {
    float4 v = row[t];
    s += v.x * v.x + v.y * v.y + v.z * v.z + v.w * v.w;
  }
  enorm[k] = s;
}

// ---------------- kernel 1: WMMA distance GEMM + argmin ----------------
// block = 128 threads (4 wave32); wave w owns n-tile (blockIdx.x*4 + w), 16 rows.
// All waves share one staged 16x64 E tile in LDS per k-iteration.
__global__ void __launch_bounds__(128) vq_argmin_kernel(const float* __restrict__ X,
                                                        const float* __restrict__ E,
                                                        const float* __restrict__ enorm,
                                                        unsigned* __restrict__ idxout) {
  __shared__ float ldsE[16 * LDS_STRIDE];

  const int tid   = threadIdx.x;
  const int lane  = tid & 31;
  const int wave  = tid >> 5;
  const int nbase = (blockIdx.x * 4 + wave) * 16;      // first latent row of this wave's tile
  const int bidx  = nbase >> 10;                       // batch index (16 | 1024 -> uniform)
  const int hw0   = nbase & (HW - 1);
  const int col   = lane & 15;                         // WMMA column / row-in-tile selector
  const int hisel = (lane >> 4) << 1;                  // 0 for lanes 0-15, 2 for lanes 16-31

  // X element for row (nbase+col), channel d:  bidx*65536 + d*1024 + hw0 + col
  const int xbase = bidx * (DD * HW) + hw0 + col;

  // ---- preload the whole 16x64 A tile as 16 f32 WMMA fragments (reused for all k) ----
  // A 16x4 layout: VGPR0 = K0 | K2 (lane halves), VGPR1 = K1 | K3.
  v2f a[16];
#pragma unroll
  for (int t = 0; t < 16; ++t) {
    int dd = 4 * t + hisel;
    a[t].x = X[xbase + dd * HW];
    a[t].y = X[xbase + (dd + 1) * HW];
  }

  float    minv[8];
  unsigned mini[8];
#pragma unroll
  for (int j = 0; j < 8; ++j) { minv[j] = __builtin_inff(); mini[j] = 0u; }

  for (int kb = 0; kb < KK; kb += 16) {
    __syncthreads();                                   // protect previous tile's readers
    // stage contiguous 4KB E tile [kb..kb+16) x [0..64) with padded stride
    for (int i = tid; i < 16 * DD; i += 128) {
      int k = i >> 6, d = i & 63;
      ldsE[k * LDS_STRIDE + d] = E[(size_t)kb * DD + i];
    }
    __syncthreads();

    float en = enorm[kb + col];

    v8f c = {0.f, 0.f, 0.f, 0.f, 0.f, 0.f, 0.f, 0.f};
#pragma unroll
    for (int t = 0; t < 16; ++t) {
      int dd = 4 * t + hisel;
      // B 4x16 layout mirrors A: VGPR0 = K0|K2, VGPR1 = K1|K3; B[krow][ncol]=E[kb+ncol][d0+krow]
      v2f bf;
      bf.x = ldsE[col * LDS_STRIDE + dd];
      bf.y = ldsE[col * LDS_STRIDE + dd + 1];
      c = __builtin_amdgcn_wmma_f32_16x16x4_f32(false, a[t], false, bf,
                                                (short)0, c, false, false);
    }

    // dist(row, kb+col) = ||e||^2 - 2*dot  (||x||^2 is row-constant, irrelevant to argmin)
    unsigned kidx = (unsigned)(kb + col);
#pragma unroll
    for (int j = 0; j < 8; ++j) {
      float dist = fmaf(-2.f, c[j], en);
      if (dist < minv[j]) { minv[j] = dist; mini[j] = kidx; }   // strict < keeps first k
    }
  }

  // ---- argmin reduction across the 16 column-lanes of each half-wave ----
  // C/D layout: lanes 0-15 hold rows 0-7 (VGPR j -> M=j), lanes 16-31 hold rows 8-15.
#pragma unroll
  for (int j = 0; j < 8; ++j) {
    float    mv = minv[j];
    unsigned mi = mini[j];
#pragma unroll
    for (int m = 1; m < 16; m <<= 1) {
      float    ov = __shfl_xor(mv, m, 32);
      unsigned oi = __shfl_xor(mi, m, 32);
      if (ov < mv || (ov == mv && oi < mi)) { mv = ov; mi = oi; }  // first-occurrence ties
    }
    minv[j] = mv; mini[j] = mi;
  }

  if (col == 0) {
    int rbase = (lane >= 16) ? 8 : 0;
#pragma unroll
    for (int j = 0; j < 8; ++j) idxout[nbase + rbase + j] = mini[j];
  }
}

// ---------------- kernel 2: gather quant (NCHW) + per-block partial loss ----------------
__global__ void __launch_bounds__(256) vq_gather_kernel(const float* __restrict__ X,
                                                        const float* __restrict__ E,
                                                        const unsigned* __restrict__ idx,
                                                        float* __restrict__ out,
                                                        float* __restrict__ partials) {
  const int tid = threadIdx.x;
  const int i   = blockIdx.x * 256 + tid;              // flat NCHW index
  const int b   = i >> 16;                             // / (D*HW)
  const int d   = (i >> 10) & 63;
  const int hw  = i & (HW - 1);
  const int n   = (b << 10) | hw;

  unsigned k = idx[n];
  float q = E[(size_t)k * DD + d];                     // scattered 4B, L2-resident (256KB)
  float x = X[i];
  out[i] = q;                                          // straight-through forward == quant
  float diff = q - x;
  float val  = diff * diff;

  __shared__ float s[256];
  s[tid] = val;
  __syncthreads();
#pragma unroll
  for (int off = 128; off > 0; off >>= 1) {
    if (tid < off) s[tid] += s[tid + off];
    __syncthreads();
  }
  if (tid == 0) partials[blockIdx.x] = s[0];
}

// ---------------- kernel 3: final deterministic loss reduction ----------------
__global__ void __launch_bounds__(256) vq_loss_kernel(const float* __restrict__ partials,
                                                      int n, float* __restrict__ loss) {
  const int tid = threadIdx.x;
  float acc = 0.f;
  for (int i = tid; i < n; i += 256) acc += partials[i];
  __shared__ float s[256];
  s[tid] = acc;
  __syncthreads();
#pragma unroll
  for (int off = 128; off > 0; off >>= 1) {
    if (tid < off) s[tid] += s[tid + off];
    __syncthreads();
  }
  // vq_loss = (1 + beta) * mean(diff^2) = 1.25 * sum / 4194304
  if (tid == 0) *loss = s[0] * (1.25f / (float)NELEM);
}

extern "C" void kernel_launch(void* const* d_in, const int* in_sizes, int n_in,
                              void* d_out, int out_size, void* d_ws, size_t ws_size,
                              hipStream_t stream) {
  const float* lat = (const float*)d_in[0];   // [64,64,32,32] fp32
  const float* emb = (const float*)d_in[1];   // [1024,64]     fp32
  float* out = (float*)d_out;                 // quant (4194304) ++ loss (1)

  unsigned char* ws = (unsigned char*)d_ws;
  unsigned* idx      = (unsigned*)ws;                              // 65536 * 4B
  float*    enorm    = (float*)(ws + (size_t)NN * 4);              // 1024  * 4B
  float*    partials = (float*)(ws + (size_t)NN * 4 + KK * 4);     // 16384 * 4B

  vq_enorm_kernel <<<KK / 256, 256, 0, stream>>>(emb, enorm);
  vq_argmin_kernel<<<NN / 64,  128, 0, stream>>>(lat, emb, enorm, idx);
  vq_gather_kernel<<<NELEM / 256, 256, 0, stream>>>(lat, emb, idx, out, partials);
  vq_loss_kernel  <<<1, 256, 0, stream>>>(partials, NELEM / 256, out + NELEM);
}